// SoftMoELayer_23819888624292
// MI455X (gfx1250) — compile-verified
//
#include <hip/hip_runtime.h>
#include <hip/hip_bf16.h>

// ---------------------------------------------------------------------------
// SoftMoE forward for MI455X (gfx1250, wave32, f32 WMMA 16x16x4)
//
// Shapes: B=4 T=2048 D=1024  N_EXPERTS=32 N_SLOTS=2 D_FF=4096
// Bandwidth bound: w1+w2 = 1.07 GB f32 per call -> ~46us floor @ 23.3 TB/s.
// ---------------------------------------------------------------------------

typedef __attribute__((ext_vector_type(2))) float v2f;
typedef __attribute__((ext_vector_type(8))) float v8f;

#define NTOK   8192          // B*T
#define DIM    1024          // D
#define NSLOT  64            // N_EXPERTS * N_SLOTS
#define NEXP   32
#define DFF    4096
#define KSPLIT 8             // split-K factor for x_hat GEMM

// D = A(16x4,f32) * B(4x16,f32) + C(16x16,f32)
#define WMMA_F32(a, b, c) \
  __builtin_amdgcn_wmma_f32_16x16x4_f32(false, (a), false, (b), (short)0, (c), false, false)

__device__ __forceinline__ float gelu_tanh(float x) {
  const float k0 = 0.7978845608028654f;  // sqrt(2/pi)
  const float k1 = 0.044715f;
  float x3 = x * x * x;
  return 0.5f * x * (1.0f + tanhf(k0 * (x + k1 * x3)));
}

// ---------------------------------------------------------------------------
// 1) logits: mat[8192,64] = toks[8192,1024] @ W[1024,64]
//    one wave per 16x16 tile; 512 mtiles x 4 ntiles = 2048 waves
// ---------------------------------------------------------------------------
__global__ __launch_bounds__(256) void k_logits(const float* __restrict__ toks,
                                                const float* __restrict__ W,
                                                float* __restrict__ mat) {
  const int wave = (blockIdx.x * blockDim.x + threadIdx.x) >> 5;
  const int lane = threadIdx.x & 31;
  const int l16 = lane & 15, half = lane >> 4;
  const int mtile = wave >> 2;       // 0..511
  const int ntile = wave & 3;        // 0..3

  const float* arow = toks + (mtile * 16 + l16) * DIM;   // A row (M = l16)
  const float* bcol = W + ntile * 16 + l16;              // B col (N = l16)

  v8f c = {};
  for (int k = 0; k < DIM; k += 4) {
    const int kk = k + 2 * half;                         // half-wave K pair
    v2f a; a.x = arow[kk];            a.y = arow[kk + 1];
    v2f b; b.x = bcol[kk * NSLOT];    b.y = bcol[(kk + 1) * NSLOT];
    c = WMMA_F32(a, b, c);
  }
  // C layout: VGPR r -> M = r + 8*half, N = l16
  float* out = mat + (mtile * 16 + half * 8) * NSLOT + ntile * 16 + l16;
#pragma unroll
  for (int r = 0; r < 8; ++r) out[r * NSLOT] = c[r];
}

// ---------------------------------------------------------------------------
// 2) dispatch softmax stats: per-column (slot) max and 1/sum(exp) over tokens
// ---------------------------------------------------------------------------
__global__ __launch_bounds__(256) void k_colstats(const float* __restrict__ mat,
                                                  float* __restrict__ colmax,
                                                  float* __restrict__ colrcp) {
  const int c = blockIdx.x;          // 0..63
  const int t = threadIdx.x;
  __shared__ float red[256];

  float m = -3.0e38f;
  for (int r = t; r < NTOK; r += 256) m = fmaxf(m, mat[r * NSLOT + c]);
  red[t] = m;
  __syncthreads();
  for (int s = 128; s > 0; s >>= 1) {
    if (t < s) red[t] = fmaxf(red[t], red[t + s]);
    __syncthreads();
  }
  const float cm = red[0];
  __syncthreads();

  float sum = 0.0f;
  for (int r = t; r < NTOK; r += 256) sum += __expf(mat[r * NSLOT + c] - cm);
  red[t] = sum;
  __syncthreads();
  for (int s = 128; s > 0; s >>= 1) {
    if (t < s) red[t] += red[t + s];
    __syncthreads();
  }
  if (t == 0) { colmax[c] = cm; colrcp[c] = 1.0f / red[0]; }
}

// ---------------------------------------------------------------------------
// 3) probabilities: one wave per token row (lane handles cols l and l+32)
//    col_probs = softmax over the 64 slots (combine)
//    row_probs = exp(mat - colmax) * colrcp      (dispatch)
// ---------------------------------------------------------------------------
__global__ __launch_bounds__(256) void k_probs(const float* __restrict__ mat,
                                               const float* __restrict__ colmax,
                                               const float* __restrict__ colrcp,
                                               float* __restrict__ row_probs,
                                               float* __restrict__ col_probs) {
  const int row = (blockIdx.x * blockDim.x + threadIdx.x) >> 5;
  const int lane = threadIdx.x & 31;
  const float m0 = mat[row * NSLOT + lane];
  const float m1 = mat[row * NSLOT + lane + 32];

  float rmax = fmaxf(m0, m1);
  for (int off = 16; off; off >>= 1) rmax = fmaxf(rmax, __shfl_xor(rmax, off));
  const float e0 = __expf(m0 - rmax), e1 = __expf(m1 - rmax);
  float rsum = e0 + e1;
  for (int off = 16; off; off >>= 1) rsum += __shfl_xor(rsum, off);
  const float rr = 1.0f / rsum;

  col_probs[row * NSLOT + lane]      = e0 * rr;
  col_probs[row * NSLOT + lane + 32] = e1 * rr;
  row_probs[row * NSLOT + lane]      = __expf(m0 - colmax[lane]) * colrcp[lane];
  row_probs[row * NSLOT + lane + 32] = __expf(m1 - colmax[lane + 32]) * colrcp[lane + 32];
}

// ---------------------------------------------------------------------------
// 4) x_hat partials: x_hat[64,1024] = row_probs^T @ toks, split-K x8
//    wave id = (kchunk*4 + mtile)*64 + ntile  -> 2048 waves
// ---------------------------------------------------------------------------
__global__ __launch_bounds__(256) void k_xhat(const float* __restrict__ row_probs,
                                              const float* __restrict__ toks,
                                              float* __restrict__ xhat_part) {
  const int wave = (blockIdx.x * blockDim.x + threadIdx.x) >> 5;
  const int lane = threadIdx.x & 31;
  const int l16 = lane & 15, half = lane >> 4;
  const int ntile = wave & 63;
  const int mtile = (wave >> 6) & 3;
  const int kchunk = wave >> 8;      // 0..7

  // A[s][t] = row_probs[t][s] (transposed access), M = slot, K = token
  const float* acol = row_probs + mtile * 16 + l16;   // + t*64
  const float* bcol = toks + ntile * 16 + l16;        // + t*1024

  v8f c = {};
  const int k0 = kchunk * (NTOK / KSPLIT);
  for (int k = k0; k < k0 + NTOK / KSPLIT; k += 4) {
    const int kk = k + 2 * half;
    v2f a; a.x = acol[kk * NSLOT];  a.y = acol[(kk + 1) * NSLOT];
    v2f b; b.x = bcol[kk * DIM];    b.y = bcol[(kk + 1) * DIM];
    c = WMMA_F32(a, b, c);
  }
  float* out = xhat_part + kchunk * (NSLOT * DIM)
             + (mtile * 16 + half * 8) * DIM + ntile * 16 + l16;
#pragma unroll
  for (int r = 0; r < 8; ++r) out[r * DIM] = c[r];
}

__global__ __launch_bounds__(256) void k_xhat_reduce(const float* __restrict__ part,
                                                     float* __restrict__ xhat) {
  const int i = blockIdx.x * blockDim.x + threadIdx.x;  // 0..65535
  float s = 0.0f;
#pragma unroll
  for (int j = 0; j < KSPLIT; ++j) s += part[j * (NSLOT * DIM) + i];
  xhat[i] = s;
}

// ---------------------------------------------------------------------------
// 5) expert MLP layer 1: h[e, p, :] = gelu(xhat[e,p,:] @ w1[e] + b1[e])
//    M padded 2->16 (rows 2..15 fed zeros). 32 experts x 256 ntiles = 8192 waves.
//    This kernel streams the 512 MB w1 once -> bandwidth-critical.
// ---------------------------------------------------------------------------
__global__ __launch_bounds__(256) void k_expert1(const float* __restrict__ xhat,
                                                 const float* __restrict__ w1,
                                                 const float* __restrict__ b1,
                                                 float* __restrict__ h) {
  const int wave = (blockIdx.x * blockDim.x + threadIdx.x) >> 5;
  const int lane = threadIdx.x & 31;
  const int l16 = lane & 15, half = lane >> 4;
  const int e = wave >> 8;           // expert
  const int ntile = wave & 255;      // DFF/16 tiles

  const float* arow = xhat + (size_t)(e * 2 + (l16 < 2 ? l16 : 0)) * DIM;
  const bool avalid = (l16 < 2);
  const float* bcol = w1 + (size_t)e * DIM * DFF + ntile * 16 + l16;

  v8f c = {};
  for (int k = 0; k < DIM; k += 4) {
    const int kk = k + 2 * half;
    v2f a;
    a.x = avalid ? arow[kk] : 0.0f;
    a.y = avalid ? arow[kk + 1] : 0.0f;
    v2f b;
    b.x = bcol[(size_t)kk * DFF];
    b.y = bcol[(size_t)(kk + 1) * DFF];
    c = WMMA_F32(a, b, c);
  }
  // valid rows M=0 (c[0], lanes 0-15) and M=1 (c[1], lanes 0-15)
  if (lane < 16) {
    const int col = ntile * 16 + lane;
    const float bb = b1[e * DFF + col];
    h[(size_t)(e * 2 + 0) * DFF + col] = gelu_tanh(c[0] + bb);
    h[(size_t)(e * 2 + 1) * DFF + col] = gelu_tanh(c[1] + bb);
  }
}

// ---------------------------------------------------------------------------
// 6) expert MLP layer 2: y_hat[e, p, :] = h[e,p,:] @ w2[e] + b2[e]
//    32 experts x 64 ntiles = 2048 waves; streams the 512 MB w2 once.
// ---------------------------------------------------------------------------
__global__ __launch_bounds__(256) void k_expert2(const float* __restrict__ h,
                                                 const float* __restrict__ w2,
                                                 const float* __restrict__ b2,
                                                 float* __restrict__ y_hat) {
  const int wave = (blockIdx.x * blockDim.x + threadIdx.x) >> 5;
  const int lane = threadIdx.x & 31;
  const int l16 = lane & 15, half = lane >> 4;
  const int e = wave >> 6;           // expert
  const int ntile = wave & 63;       // DIM/16 tiles

  const float* arow = h + (size_t)(e * 2 + (l16 < 2 ? l16 : 0)) * DFF;
  const bool avalid = (l16 < 2);
  const float* bcol = w2 + (size_t)e * DFF * DIM + ntile * 16 + l16;

  v8f c = {};
  for (int k = 0; k < DFF; k += 4) {
    const int kk = k + 2 * half;
    v2f a;
    a.x = avalid ? arow[kk] : 0.0f;
    a.y = avalid ? arow[kk + 1] : 0.0f;
    v2f b;
    b.x = bcol[(size_t)kk * DIM];
    b.y = bcol[(size_t)(kk + 1) * DIM];
    c = WMMA_F32(a, b, c);
  }
  if (lane < 16) {
    const int col = ntile * 16 + lane;
    const float bb = b2[e * DIM + col];
    y_hat[(size_t)(e * 2 + 0) * DIM + col] = c[0] + bb;
    y_hat[(size_t)(e * 2 + 1) * DIM + col] = c[1] + bb;
  }
}

// ---------------------------------------------------------------------------
// 7) combine: y[8192,1024] = col_probs[8192,64] @ y_hat[64,1024]
//    512 mtiles x 64 ntiles = 32768 waves; K = 64 -> 16 WMMA per wave
// ---------------------------------------------------------------------------
__global__ __launch_bounds__(256) void k_combine(const float* __restrict__ col_probs,
                                                 const float* __restrict__ y_hat,
                                                 float* __restrict__ out) {
  const int wave = (blockIdx.x * blockDim.x + threadIdx.x) >> 5;
  const int lane = threadIdx.x & 31;
  const int l16 = lane & 15, half = lane >> 4;
  const int ntile = wave & 63;
  const int mtile = wave >> 6;       // 0..511

  const float* arow = col_probs + (mtile * 16 + l16) * NSLOT;
  const float* bcol = y_hat + ntile * 16 + l16;

  v8f c = {};
  for (int k = 0; k < NSLOT; k += 4) {
    const int kk = k + 2 * half;
    v2f a; a.x = arow[kk];          a.y = arow[kk + 1];
    v2f b; b.x = bcol[kk * DIM];    b.y = bcol[(kk + 1) * DIM];
    c = WMMA_F32(a, b, c);
  }
  float* o = out + (mtile * 16 + half * 8) * DIM + ntile * 16 + l16;
#pragma unroll
  for (int r = 0; r < 8; ++r) o[r * DIM] = c[r];
}

// ---------------------------------------------------------------------------
// launch
// ---------------------------------------------------------------------------
extern "C" void kernel_launch(void* const* d_in, const int* in_sizes, int n_in,
                              void* d_out, int out_size, void* d_ws, size_t ws_size,
                              hipStream_t stream) {
  const float* toks = (const float*)d_in[0];  // [8192,1024]
  const float* W    = (const float*)d_in[1];  // [1024,64]
  const float* w1   = (const float*)d_in[2];  // [32,1024,4096]
  const float* b1   = (const float*)d_in[3];  // [32,4096]
  const float* w2   = (const float*)d_in[4];  // [32,4096,1024]
  const float* b2   = (const float*)d_in[5];  // [32,1024]
  float* out = (float*)d_out;                 // [8192,1024]

  // workspace layout (floats)
  float* ws        = (float*)d_ws;
  float* mat       = ws;                               // 524288
  float* row_probs = mat + NTOK * NSLOT;               // 524288
  float* col_probs = row_probs + NTOK * NSLOT;         // 524288
  float* colmax    = col_probs + NTOK * NSLOT;         // 64
  float* colrcp    = colmax + NSLOT;                   // 64
  float* xhat      = colrcp + NSLOT;                   // 65536
  float* h         = xhat + NSLOT * DIM;               // 262144
  float* y_hat     = h + NSLOT * DFF;                  // 65536
  float* xhat_part = y_hat + NSLOT * DIM;              // 524288

  // 1) logits: 2048 waves -> 256 blocks x 256 threads
  k_logits<<<256, 256, 0, stream>>>(toks, W, mat);
  // 2) dispatch-softmax column stats
  k_colstats<<<NSLOT, 256, 0, stream>>>(mat, colmax, colrcp);
  // 3) probabilities: 8192 waves -> 1024 blocks
  k_probs<<<1024, 256, 0, stream>>>(mat, colmax, colrcp, row_probs, col_probs);
  // 4) x_hat split-K partials: 2048 waves -> 256 blocks, then reduce
  k_xhat<<<256, 256, 0, stream>>>(row_probs, toks, xhat_part);
  k_xhat_reduce<<<(NSLOT * DIM) / 256, 256, 0, stream>>>(xhat_part, xhat);
  // 5) expert layer 1: 8192 waves -> 1024 blocks (streams 512 MB w1)
  k_expert1<<<1024, 256, 0, stream>>>(xhat, w1, b1, h);
  // 6) expert layer 2: 2048 waves -> 256 blocks (streams 512 MB w2)
  k_expert2<<<256, 256, 0, stream>>>(h, w2, b2, y_hat);
  // 7) combine: 32768 waves -> 4096 blocks
  k_combine<<<4096, 256, 0, stream>>>(col_probs, y_hat, out);
}